// GNN_EncDec_85005992722725
// MI455X (gfx1250) — compile-verified
//
#include <hip/hip_runtime.h>
#include <math.h>

typedef __attribute__((ext_vector_type(2))) float v2f;
typedef __attribute__((ext_vector_type(8))) float v8f;
typedef __attribute__((ext_vector_type(4))) unsigned int v4u;
typedef __attribute__((ext_vector_type(4))) int v4i;
typedef __attribute__((ext_vector_type(8))) int v8i;

// ---------------------------------------------------------------------------
// deg[i] = 1.0 (self loop weight)
__global__ void k_deg_init(float* __restrict__ deg, int n) {
    int i = blockIdx.x * blockDim.x + threadIdx.x;
    if (i < n) deg[i] = 1.0f;
}

// deg[dst[e]] += w[e]
__global__ void k_deg_accum(float* __restrict__ deg, const int* __restrict__ dst,
                            const float* __restrict__ w, int E) {
    int e = blockIdx.x * blockDim.x + threadIdx.x;
    if (e < E) atomicAdd(&deg[dst[e]], w[e]);
}

// in-place: deg -> dinv = rsqrt(deg)
__global__ void k_dinv(float* __restrict__ deg, int n) {
    int i = blockIdx.x * blockDim.x + threadIdx.x;
    if (i < n) {
        float v = deg[i];
        deg[i] = (v > 0.0f) ? rsqrtf(v) : 0.0f;
    }
}

// norm[e] = dinv[src]*w*dinv[dst]
__global__ void k_norm(float* __restrict__ norm, const float* __restrict__ dinv,
                       const int* __restrict__ src, const int* __restrict__ dst,
                       const float* __restrict__ w, int E) {
    int e = blockIdx.x * blockDim.x + threadIdx.x;
    if (e < E) norm[e] = dinv[src[e]] * w[e] * dinv[dst[e]];
}

// ---------------------------------------------------------------------------
// C[M,N] = A[M,K] @ B[K,N] with V_WMMA_F32_16X16X4_F32.
// Block = 256 threads = 8 waves; one block computes a 16(M) x 128(N) slab.
// The shared 16xK A tile (contiguous rows!) is DMA'd into LDS once per block
// by the Tensor Data Mover, then each wave computes one 16x16 tile of C.
// Fragment layouts per ISA 7.12.2:
//   A 16x4 : lane<16 -> K=k,k+1 ; lane>=16 -> K=k+2,k+3
//   B 4x16 : VGPR v, lanes0-15 row K=k+v, lanes16-31 row K=k+v+2
//   C      : VGPR r, lanes0-15 row m0+r,  lanes16-31 row m0+8+r
__global__ __launch_bounds__(256) void k_gemm_wmma(
        const float* __restrict__ A, const float* __restrict__ B,
        float* __restrict__ C, int M, int K, int N) {
    __shared__ float ldsA[16 * 256];            // K <= 256 -> up to 16 KB

    const int wave = threadIdx.x >> 5;
    const int lane = threadIdx.x & 31;
    const int half = lane >> 4;                 // lane-half select
    const int lm   = lane & 15;
    const int m0 = blockIdx.x * 16;
    const int n0 = wave * 16;

    const unsigned int L = 16u * (unsigned int)K;            // tile elements
    const unsigned long long rem = (unsigned long long)(M - m0) * (unsigned int)K;
    const unsigned int td0 = (rem < (unsigned long long)L) ? (unsigned int)rem : L;

#if __has_builtin(__builtin_amdgcn_tensor_load_to_lds) && \
    __has_builtin(__builtin_amdgcn_s_wait_tensorcnt)
    if (threadIdx.x < 32) {                     // wave 0 issues the TDM DMA
        const unsigned long long ga =
            (unsigned long long)(uintptr_t)(A + (size_t)m0 * K);
        const unsigned int lds_off = (unsigned int)(uintptr_t)&ldsA[0];

        // D# group 0: count=1 | lds_addr | global_addr[56:0] | type=2
        v4u g0;
        g0.x = 0x1u;
        g0.y = lds_off;
        g0.z = (unsigned int)ga;
        g0.w = ((unsigned int)(ga >> 32) & 0x01FFFFFFu) | 0x80000000u;

        // D# group 1: data_size=2 (4B), tensor_dim0=td0 (zeros past end),
        // tensor_dim1=1, tile_dim0=L, tile_dim1=1, dim0_stride=L
        v8i g1;
        g1[0] = 0x00020000;                          // wg_mask=0, data_size=4B
        g1[1] = (int)((td0 & 0xFFFFu) << 16);        // tensor_dim0[15:0]
        g1[2] = (int)((td0 >> 16) | 0x00010000u);    // tensor_dim0[31:16] | tensor_dim1=1
        g1[3] = (int)((L & 0xFFFFu) << 16);          // tile_dim0
        g1[4] = 0x00000001;                          // tile_dim1=1, tile_dim2=0
        g1[5] = (int)L;                              // tensor_dim0_stride[31:0]
        g1[6] = 0;
        g1[7] = 0;

        v4i gz4 = {0, 0, 0, 0};                      // groups 2/3 unused (<=2D)
        v8i gz8 = {0, 0, 0, 0, 0, 0, 0, 0};          // extra group (unused)
        __builtin_amdgcn_tensor_load_to_lds(g0, g1, gz4, gz4, gz8, 0);
        __builtin_amdgcn_s_wait_tensorcnt(0);
    }
#else
    for (unsigned int idx = threadIdx.x; idx < L; idx += 256)
        ldsA[idx] = (idx < td0) ? A[(size_t)m0 * K + idx] : 0.0f;
#endif
    __syncthreads();

    if (n0 < N) {                               // wave-uniform: EXEC all-ones
        const float* __restrict__ Arow = &ldsA[lm * K];
        const float* __restrict__ Bcol = B + n0 + lm;

        v8f acc = {};
        #pragma unroll 8
        for (int k = 0; k < K; k += 4) {
            const int ka = k + half * 2;
            v2f a;
            a.x = Arow[ka];
            a.y = Arow[ka + 1];
            v2f b;
            b.x = Bcol[(size_t)ka * N];
            b.y = Bcol[(size_t)(ka + 1) * N];
            acc = __builtin_amdgcn_wmma_f32_16x16x4_f32(
                /*neg_a=*/false, a, /*neg_b=*/false, b,
                /*c_mod=*/(short)0, acc, /*reuse_a=*/false, /*reuse_b=*/false);
        }

        #pragma unroll
        for (int r = 0; r < 8; ++r) {
            const int row = m0 + half * 8 + r;
            if (row < M) C[(size_t)row * N + n0 + lm] = acc[r];
        }
    }
}

// ---------------------------------------------------------------------------
// agg[i,:] = h[i,:] * dinv[i]^2   (self-loop message; also initializes agg)
__global__ void k_self_init(float* __restrict__ agg, const float* __restrict__ h,
                            const float* __restrict__ dinv, int n, int hid) {
    const int quads = hid >> 2;
    int t = blockIdx.x * blockDim.x + threadIdx.x;
    if (t >= n * quads) return;
    const int i = t / quads;
    const float di = dinv[i];
    const float c = di * di;
    float4 hv = ((const float4*)h)[t];
    float4 r;
    r.x = hv.x * c; r.y = hv.y * c; r.z = hv.z * c; r.w = hv.w * c;
    ((float4*)agg)[t] = r;
}

// agg[dst[e],:] += h[src[e],:] * norm[e]   (one wave32 per edge, float4/lane)
__global__ void k_scatter(float* __restrict__ agg, const float* __restrict__ h,
                          const float* __restrict__ norm,
                          const int* __restrict__ src, const int* __restrict__ dst,
                          int E, int hid) {
    int t = blockIdx.x * blockDim.x + threadIdx.x;
    if (t >= E * 32) return;
    const int e = t >> 5;
    const int q = t & 31;                       // hid=128 -> 32 quads per row
    const int s = src[e];
    const int d = dst[e];
    const float nm = norm[e];
    float4 hv = ((const float4*)(h + (size_t)s * hid))[q];
    float* ag = agg + (size_t)d * hid + q * 4;
    atomicAdd(ag + 0, hv.x * nm);
    atomicAdd(ag + 1, hv.y * nm);
    atomicAdd(ag + 2, hv.z * nm);
    atomicAdd(ag + 3, hv.w * nm);
}

// out = gelu(in + bias)   (exact erf GELU, float4)
__global__ void k_bias_gelu(float* __restrict__ out, const float* __restrict__ in,
                            const float* __restrict__ bias, int n, int hid) {
    const int quads = hid >> 2;
    int t = blockIdx.x * blockDim.x + threadIdx.x;
    if (t >= n * quads) return;
    const int q = t % quads;
    float4 v = ((const float4*)in)[t];
    float4 bb = ((const float4*)bias)[q];
    float x0 = v.x + bb.x, x1 = v.y + bb.y, x2 = v.z + bb.z, x3 = v.w + bb.w;
    const float is2 = 0.7071067811865475f;
    float4 r;
    r.x = 0.5f * x0 * (1.0f + erff(x0 * is2));
    r.y = 0.5f * x1 * (1.0f + erff(x1 * is2));
    r.z = 0.5f * x2 * (1.0f + erff(x2 * is2));
    r.w = 0.5f * x3 * (1.0f + erff(x3 * is2));
    ((float4*)out)[t] = r;
}

// ---------------------------------------------------------------------------
extern "C" void kernel_launch(void* const* d_in, const int* in_sizes, int n_in,
                              void* d_out, int out_size, void* d_ws, size_t ws_size,
                              hipStream_t stream) {
    const float* x   = (const float*)d_in[0];
    const int*   ei  = (const int*)  d_in[1];
    const float* ew  = (const float*)d_in[2];
    const float* W1  = (const float*)d_in[3];
    const float* b1  = (const float*)d_in[4];
    const float* W2  = (const float*)d_in[5];
    const float* b2  = (const float*)d_in[6];
    float* out = (float*)d_out;

    const int hid    = in_sizes[4];              // 128
    const int in_dim = in_sizes[3] / hid;        // 256
    const int n      = in_sizes[0] / in_dim;     // 100000
    const int E      = in_sizes[2];              // 1600000
    const int*   src = ei;
    const int*   dst = ei + E;

    // workspace layout (256B aligned slabs)
    size_t off = 0;
    auto take = [&](size_t bytes) {
        size_t p = off;
        off += (bytes + 255) & ~(size_t)255;
        return p;
    };
    char* ws = (char*)d_ws;
    float* dinv = (float*)(ws + take((size_t)n * 4));        // deg -> dinv
    float* nrm  = (float*)(ws + take((size_t)E * 4));        // per-edge norm
    float* bufA = (float*)(ws + take((size_t)n * hid * 4));  // h1, then g1
    float* bufB = (float*)(ws + take((size_t)n * hid * 4));  // agg1, then h2
    (void)ws_size; (void)n_in; (void)out_size;

    const int T = 256;
    const int nBlk  = (n + T - 1) / T;
    const int eBlk  = (E + T - 1) / T;
    const int fQuads = n * (hid / 4);
    const int fBlk  = (fQuads + T - 1) / T;
    const int sBlk  = (E * 32 + T - 1) / T;
    const int mTiles = (n + 15) / 16;

    // normalization coefficients
    k_deg_init <<<nBlk, T, 0, stream>>>(dinv, n);
    k_deg_accum<<<eBlk, T, 0, stream>>>(dinv, dst, ew, E);
    k_dinv     <<<nBlk, T, 0, stream>>>(dinv, n);
    k_norm     <<<eBlk, T, 0, stream>>>(nrm, dinv, src, dst, ew, E);

    // ---- layer 1: h1 = x @ W1 ; agg1 = scatter(h1*norm) ; g1 = gelu(agg1+b1)
    k_gemm_wmma<<<mTiles, T, 0, stream>>>(x, W1, bufA, n, in_dim, hid);
    k_self_init<<<fBlk, T, 0, stream>>>(bufB, bufA, dinv, n, hid);
    k_scatter  <<<sBlk, T, 0, stream>>>(bufB, bufA, nrm, src, dst, E, hid);
    k_bias_gelu<<<fBlk, T, 0, stream>>>(bufA, bufB, b1, n, hid);

    // ---- layer 2: h2 = g1 @ W2 ; agg2 (in d_out) ; out = gelu(agg2+b2)
    k_gemm_wmma<<<mTiles, T, 0, stream>>>(bufA, W2, bufB, n, hid, hid);
    k_self_init<<<fBlk, T, 0, stream>>>(out, bufB, dinv, n, hid);
    k_scatter  <<<sBlk, T, 0, stream>>>(out, bufB, nrm, src, dst, E, hid);
    k_bias_gelu<<<fBlk, T, 0, stream>>>(out, out, b2, n, hid);
}